// DynamicTemporalEncoder_56470230008108
// MI455X (gfx1250) — compile-verified
//
#include <hip/hip_runtime.h>
#include <hip/hip_bf16.h>
#include <math.h>

typedef __attribute__((ext_vector_type(2))) float v2f;
typedef __attribute__((ext_vector_type(8))) float v8f;

#define DT 0.005f
// sizes: B=2, C*H*W=2048 -> 4096 channels, T=64, k=16, Tp=65, S=4, E=64

// ---------------------------------------------------------------------------
// Kernel 0: combined[s*64+e][j] = (decay[s][j]/sum_j decay) * tk[s][e][j]
// also zero the 8-float variance accumulator. 1 block x 256 threads.
// ---------------------------------------------------------------------------
__global__ void k_prep(const float* __restrict__ tc, const float* __restrict__ tk,
                       float* __restrict__ comb, float* __restrict__ accum) {
    int t = threadIdx.x;            // 0..255  == se index (s = t>>6, e = t&63)
    if (t < 8) accum[t] = 0.0f;
    int s = t >> 6;
    float inv_tau = DT / tc[s];
    float d[16];
    float sum = 0.0f;
#pragma unroll
    for (int j = 0; j < 16; ++j) { d[j] = expf(-(float)j * inv_tau); sum += d[j]; }
    float inv = 1.0f / sum;
#pragma unroll
    for (int j = 0; j < 16; ++j)
        comb[t * 16 + j] = d[j] * inv * tk[t * 16 + j];
}

// ---------------------------------------------------------------------------
// Kernel 1: per-channel im2col GEMM X[64x16] @ comb^T[16x256] via WMMA f32
// 16x16x4 with FOUR parallel accumulators (one per 16-row m-tile) so every
// B fragment feeds 4 independent back-to-back WMMAs; row t=64 via scalar
// FMAs. Unbiased variance per (s,e) column; LDS -> 4 global atomics/block.
// grid 512, block 256 (1 channel per wave).
// ---------------------------------------------------------------------------
__global__ __launch_bounds__(256) void k_stats(const float* __restrict__ events,
                                               const float* __restrict__ comb,
                                               float* __restrict__ accum) {
    __shared__ float sComb[256 * 17];   // row stride 17: bank-conflict-free
    __shared__ float sXp[8][80];        // zero-padded x per wave
    __shared__ float sAcc[4];

    int tid = threadIdx.x;
    int lane = tid & 31, w = tid >> 5;
#pragma unroll
    for (int j = 0; j < 16; ++j) sComb[tid * 17 + j] = comb[tid * 16 + j];
    if (tid < 4) sAcc[tid] = 0.0f;

    int ch = blockIdx.x * 8 + w;                 // 0..4095, b uniform per block
    const float* x = events + (size_t)ch * 64;
    // xp[i] = x[i-8] for 8<=i<72 else 0  (im2col: X[t][j] = xp[t+j])
    sXp[w][lane]      = (lane >= 8) ? x[lane - 8] : 0.0f;
    sXp[w][32 + lane] = x[24 + lane];
    if (lane < 16) sXp[w][64 + lane] = (lane < 8) ? x[56 + lane] : 0.0f;
    __syncthreads();

    const int m    = lane & 15;                  // row/col within tile
    const int koff = (lane < 16) ? 0 : 2;        // A/B fragment K offset
    const bool lowHalf = (lane < 16);
    const float* xrow = &sXp[w][0];

    // A fragments are n-tile invariant: preload all 4 m-tiles x 4 k-steps.
    v2f afr[4][4];
#pragma unroll
    for (int mt = 0; mt < 4; ++mt)
#pragma unroll
        for (int ks = 0; ks < 4; ++ks) {
            int k0 = ks * 4 + koff;
            afr[mt][ks].x = xrow[mt * 16 + m + k0];
            afr[mt][ks].y = xrow[mt * 16 + m + k0 + 1];
        }

    for (int ntile = 0; ntile < 16; ++ntile) {   // 16 col-tiles of 16 (s,e)
        const float* brow = &sComb[(ntile * 16 + m) * 17];
        v8f c0 = {}, c1 = {}, c2 = {}, c3 = {};
#pragma unroll
        for (int ks = 0; ks < 4; ++ks) {
            int k0 = ks * 4 + koff;
            v2f bv;
            bv.x = brow[k0];
            bv.y = brow[k0 + 1];
            c0 = __builtin_amdgcn_wmma_f32_16x16x4_f32(false, afr[0][ks], false, bv, (short)0, c0, false, false);
            c1 = __builtin_amdgcn_wmma_f32_16x16x4_f32(false, afr[1][ks], false, bv, (short)0, c1, false, false);
            c2 = __builtin_amdgcn_wmma_f32_16x16x4_f32(false, afr[2][ks], false, bv, (short)0, c2, false, false);
            c3 = __builtin_amdgcn_wmma_f32_16x16x4_f32(false, afr[3][ks], false, bv, (short)0, c3, false, false);
        }
        float sum = 0.0f, sq = 0.0f;
#pragma unroll
        for (int v = 0; v < 8; ++v) {
            float f0 = c0[v], f1 = c1[v], f2 = c2[v], f3 = c3[v];
            sum += f0 + f1 + f2 + f3;
            sq  += f0 * f0 + f1 * f1 + f2 * f2 + f3 * f3;
        }
        // final row t=64: plain dot product (xp[72..79]==0 pads the window)
        float o64 = 0.0f;
#pragma unroll
        for (int j = 0; j < 16; ++j) o64 += xrow[64 + j] * brow[j];
        float c64 = lowHalf ? o64 : 0.0f;        // count it exactly once
        sum += c64; sq += c64 * c64;

        // combine row-halves (lanes l and l+16 hold same column, rows 0-7/8-15)
        sum += __shfl_xor(sum, 16, 32);
        sq  += __shfl_xor(sq, 16, 32);
        float var = (sq - sum * sum * (1.0f / 65.0f)) * (1.0f / 64.0f); // ddof=1
        // reduce the 16 columns of this tile (all share the same scale s)
        var += __shfl_xor(var, 8, 32);
        var += __shfl_xor(var, 4, 32);
        var += __shfl_xor(var, 2, 32);
        var += __shfl_xor(var, 1, 32);
        if (lane == 0) atomicAdd(&sAcc[ntile >> 2], var);
    }
    __syncthreads();
    if (tid < 4) {
        int b = (blockIdx.x * 8) >> 11;          // uniform per block
        atomicAdd(&accum[b * 4 + tid], sAcc[tid]);
    }
}

// ---------------------------------------------------------------------------
// Kernel 2: fv = accum/(E*Cf); h=relu(fv@w1^T+b1); attn=softmax(h@w2^T+b2);
// keff[b][e][j] = sum_s attn[b][s]*comb[s*64+e][j].  1 block x 256 threads.
// ---------------------------------------------------------------------------
__global__ void k_attn(const float* __restrict__ accum,
                       const float* __restrict__ w1, const float* __restrict__ b1,
                       const float* __restrict__ w2, const float* __restrict__ b2,
                       const float* __restrict__ comb, float* __restrict__ keff) {
    __shared__ float sAttn[2][4];
    int tid = threadIdx.x;
    if (tid < 2) {
        int b = tid;
        float fv[4];
#pragma unroll
        for (int s = 0; s < 4; ++s) fv[s] = accum[b * 4 + s] * (1.0f / (64.0f * 2048.0f));
        float h[8];
#pragma unroll
        for (int i = 0; i < 8; ++i) {
            float a = b1[i];
#pragma unroll
            for (int s = 0; s < 4; ++s) a += fv[s] * w1[i * 4 + s];
            h[i] = fmaxf(a, 0.0f);
        }
        float lg[4]; float mx = -1e30f;
#pragma unroll
        for (int s = 0; s < 4; ++s) {
            float a = b2[s];
#pragma unroll
            for (int i = 0; i < 8; ++i) a += h[i] * w2[s * 8 + i];
            lg[s] = a; mx = fmaxf(mx, a);
        }
        float den = 0.0f;
#pragma unroll
        for (int s = 0; s < 4; ++s) { lg[s] = expf(lg[s] - mx); den += lg[s]; }
        float inv = 1.0f / den;
#pragma unroll
        for (int s = 0; s < 4; ++s) sAttn[b][s] = lg[s] * inv;
    }
    __syncthreads();
#pragma unroll
    for (int i = 0; i < 8; ++i) {
        int idx = tid + i * 256;                 // 0..2047 == (b<<10)+(e<<4)+j
        int b = idx >> 10; int r = idx & 1023; int e = r >> 4; int j = r & 15;
        float acc = 0.0f;
#pragma unroll
        for (int s = 0; s < 4; ++s) acc += sAttn[b][s] * comb[(s * 64 + e) * 16 + j];
        keff[idx] = acc;
    }
}

// ---------------------------------------------------------------------------
// Kernel 3: encoded[b][e][cf][t] = (X @ keff[b]^T)[t][e] via WMMA with four
// parallel accumulators (t=0..63) + scalar row t=64; D tiles transposed
// through LDS (stride 67 => conflict-free) for coalesced stores along t.
// grid 512, block 256 (8 waves, 1 channel per wave).
// ---------------------------------------------------------------------------
__global__ __launch_bounds__(256) void k_encode(const float* __restrict__ events,
                                                const float* __restrict__ keff,
                                                float* __restrict__ out) {
    __shared__ float sK[64 * 17];
    __shared__ float sXp[8][80];
    __shared__ float sStage[8][16][67];          // [wave][e-in-tile][t(0..64)]

    int tid = threadIdx.x, lane = tid & 31, w = tid >> 5;
    int ch = blockIdx.x * 8 + w;
    int b  = ch >> 11;                           // uniform per block
    int cf = ch & 2047;

#pragma unroll
    for (int i = 0; i < 4; ++i) {                // load keff[b] (1024 floats)
        int idx = tid + i * 256;
        int e = idx >> 4, j = idx & 15;
        sK[e * 17 + j] = keff[(b << 10) + idx];
    }
    const float* x = events + (size_t)ch * 64;
    sXp[w][lane]      = (lane >= 8) ? x[lane - 8] : 0.0f;
    sXp[w][32 + lane] = x[24 + lane];
    if (lane < 16) sXp[w][64 + lane] = (lane < 8) ? x[56 + lane] : 0.0f;
    __syncthreads();

    const int m    = lane & 15;
    const int koff = (lane < 16) ? 0 : 2;
    const int moff = (lane < 16) ? 0 : 8;
    const float* xrow = &sXp[w][0];

    v2f afr[4][4];
#pragma unroll
    for (int mt = 0; mt < 4; ++mt)
#pragma unroll
        for (int ks = 0; ks < 4; ++ks) {
            int k0 = ks * 4 + koff;
            afr[mt][ks].x = xrow[mt * 16 + m + k0];
            afr[mt][ks].y = xrow[mt * 16 + m + k0 + 1];
        }

    for (int ntile = 0; ntile < 4; ++ntile) {    // 4 col-tiles of 16 e
        const float* brow = &sK[(ntile * 16 + m) * 17];
        v8f c0 = {}, c1 = {}, c2 = {}, c3 = {};
#pragma unroll
        for (int ks = 0; ks < 4; ++ks) {
            int k0 = ks * 4 + koff;
            v2f bv;
            bv.x = brow[k0];
            bv.y = brow[k0 + 1];
            c0 = __builtin_amdgcn_wmma_f32_16x16x4_f32(false, afr[0][ks], false, bv, (short)0, c0, false, false);
            c1 = __builtin_amdgcn_wmma_f32_16x16x4_f32(false, afr[1][ks], false, bv, (short)0, c1, false, false);
            c2 = __builtin_amdgcn_wmma_f32_16x16x4_f32(false, afr[2][ks], false, bv, (short)0, c2, false, false);
            c3 = __builtin_amdgcn_wmma_f32_16x16x4_f32(false, afr[3][ks], false, bv, (short)0, c3, false, false);
        }
#pragma unroll
        for (int v = 0; v < 8; ++v) {            // transpose via LDS
            sStage[w][m][ 0 + moff + v] = c0[v];
            sStage[w][m][16 + moff + v] = c1[v];
            sStage[w][m][32 + moff + v] = c2[v];
            sStage[w][m][48 + moff + v] = c3[v];
        }
        // row t=64: scalar dot; both lane-halves write the same value/addr
        float o64 = 0.0f;
#pragma unroll
        for (int j = 0; j < 16; ++j) o64 += xrow[64 + j] * brow[j];
        sStage[w][m][64] = o64;

        __asm__ volatile("s_wait_dscnt 0" ::: "memory");
        int nb = ntile * 16;
        for (int e16 = 0; e16 < 16; ++e16) {     // coalesced 65-float rows
            int e = nb + e16;
            size_t base = ((size_t)(b * 64 + e) * 2048 + cf) * 65;
#pragma unroll
            for (int t0 = 0; t0 < 96; t0 += 32) {
                int t = t0 + lane;
                if (t < 65) out[base + t] = sStage[w][e16][t];
            }
        }
        __asm__ volatile("" ::: "memory");       // keep stage reads before rewrite
    }
}

// ---------------------------------------------------------------------------
extern "C" void kernel_launch(void* const* d_in, const int* in_sizes, int n_in,
                              void* d_out, int out_size, void* d_ws, size_t ws_size,
                              hipStream_t stream) {
    const float* events = (const float*)d_in[0];   // [2,2,32,32,64]
    const float* tcs    = (const float*)d_in[1];   // [4]
    const float* tk     = (const float*)d_in[2];   // [4,64,16]
    const float* w1     = (const float*)d_in[3];   // [8,4]
    const float* b1     = (const float*)d_in[4];   // [8]
    const float* w2     = (const float*)d_in[5];   // [4,8]
    const float* b2     = (const float*)d_in[6];   // [4]
    float* out = (float*)d_out;                    // [2,64,2048,65]

    float* ws    = (float*)d_ws;
    float* comb  = ws;                             // 4096 floats
    float* accum = ws + 4096;                      // 8 floats
    float* keff  = ws + 4104;                      // 2048 floats

    k_prep  <<<1,   256, 0, stream>>>(tcs, tk, comb, accum);
    k_stats <<<512, 256, 0, stream>>>(events, comb, accum);
    k_attn  <<<1,   256, 0, stream>>>(accum, w1, b1, w2, b2, comb, keff);
    k_encode<<<512, 256, 0, stream>>>(events, keff, out);
}